// BSGRUCell_5437428596857
// MI455X (gfx1250) — compile-verified
//
#include <hip/hip_runtime.h>
#include <hip/hip_bf16.h>
#include <math.h>

// ---------------------------------------------------------------------------
// BSGRUCell on MI455X (gfx1250, wave32, WMMA)
//
// Heavy math collapses to two fixed fp32 GEMMs:
//   Gi[b,o] = input[b,:] . W2[o,:]   W2 = weight_ih flat as [3072, 512]
//   Gh[b,o] = hx[b,:]    . U2[o,:]   U2 = weight_hh flat as [3072,1024]
// then per-element routing scale + bias, then GRU gate math.
// Memory-bound: 18 MB fp32 weights/call -> V_WMMA_F32_16X16X4_F32.
// GEMM kernel is templated on (KDIM, IS_H) so all control flow is scalar.
// ---------------------------------------------------------------------------

typedef __attribute__((ext_vector_type(2))) float v2f;
typedef __attribute__((ext_vector_type(8))) float v8f;

#define BATCH   32
#define IN_SZ   512
#define HID     1024
#define NBLK    4
#define BSIZE   256      // HID / NBLK
#define G3H     3072     // 3 * HID
#define BETA    10.0f

// ---------------------------------------------------------------------------
// Kernel 1: routing softmax  ky[b,k]
// ---------------------------------------------------------------------------
__global__ __launch_bounds__(128) void bsgru_route(
    const float* __restrict__ input,   // [32,512]
    const float* __restrict__ hx,      // [32,1024]
    const float* __restrict__ wik,     // [512,4]
    const float* __restrict__ whk,     // [1024,4]
    const float* __restrict__ bik,     // [4]
    float* __restrict__ ky_ws,         // [32,4] workspace
    float* __restrict__ ky_out)        // [32,4] output tail
{
    __shared__ float logit[BATCH * NBLK];
    const int t = threadIdx.x;           // 0..127
    const int b = t >> 2;
    const int k = t & 3;

    float acc = bik[k];
    const float* ip = input + b * IN_SZ;
    const float* hp = hx + b * HID;
    #pragma unroll 8
    for (int i = 0; i < IN_SZ; ++i) acc += ip[i] * wik[i * NBLK + k];
    #pragma unroll 8
    for (int j = 0; j < HID; ++j)   acc += hp[j] * whk[j * NBLK + k];

    const float l = BETA * acc;
    logit[t] = l;
    __syncthreads();

    const float* row = &logit[b * NBLK];
    const float m = fmaxf(fmaxf(row[0], row[1]), fmaxf(row[2], row[3]));
    const float s = __expf(row[0] - m) + __expf(row[1] - m) +
                    __expf(row[2] - m) + __expf(row[3] - m);
    const float v = __expf(l - m) / s;
    ky_ws[t]  = v;
    ky_out[t] = v;
}

// ---------------------------------------------------------------------------
// Kernel 2: WMMA GEMM with routing-scale epilogue (templated: uniform ctrl).
// 192 wave-tiles per instantiation, 4 waves/block -> 48 blocks.
// Each wave computes a 32x16 output strip: two 16x16 C tiles sharing one
// B fragment per K-step of 4 (V_WMMA_F32_16X16X4_F32). K chunked by 16 so
// one WGP-scope prefetch covers 4 WMMA pairs.
// ---------------------------------------------------------------------------
template <int KDIM, bool IS_H>
__global__ __launch_bounds__(128) void bsgru_gemm(
    const float* __restrict__ A,       // [32,KDIM]  (input or hx)
    const float* __restrict__ W,       // flat [3072, KDIM] row-major
    const float* __restrict__ bias,    // [3072]
    const float* __restrict__ kx,      // [32,4] (only read when IS_H)
    const float* __restrict__ ky,      // [32,4] (workspace, from kernel 1)
    float* __restrict__ G)             // [32,3072]
{
    const int lane = threadIdx.x & 31;
    const int wave = threadIdx.x >> 5;
    const int nt   = blockIdx.x * 4 + wave;      // 0..191 (N tile)
    const int n0   = nt * 16;                    // output-column base

    // ISA 7.12.2 fragment layouts (wave32):
    //  A 16x4 f32 : lane holds M=lane&15, VGPR{0,1} = K = 2*(lane>>4) + {0,1}
    //  B  4x16 f32: lane holds N=lane&15, VGPR{0,1} = K = 2*(lane>>4) + {0,1}
    //  C 16x16 f32: VGPR r -> M = r + 8*(lane>>4), N = lane&15
    const int mrow = lane & 15;
    const int koff = (lane >> 4) * 2;
    const int ncol = n0 + (lane & 15);

    const float* __restrict__ Arow0 = A + (size_t)mrow * KDIM + koff;
    const float* __restrict__ Arow1 = A + (size_t)(mrow + 16) * KDIM + koff;
    const float* __restrict__ Wrow  = W + (size_t)ncol * KDIM + koff;

    v8f acc0 = {0.f, 0.f, 0.f, 0.f, 0.f, 0.f, 0.f, 0.f};
    v8f acc1 = {0.f, 0.f, 0.f, 0.f, 0.f, 0.f, 0.f, 0.f};

    for (int k0 = 0; k0 < KDIM; k0 += 16) {
        // pull the weight stream 4 chunks (64 floats) ahead into near caches
        __builtin_prefetch(Wrow + k0 + 64, 0, 3);
        #pragma unroll
        for (int kk = 0; kk < 16; kk += 4) {
            const v2f a0 = *(const v2f*)(Arow0 + k0 + kk);
            const v2f a1 = *(const v2f*)(Arow1 + k0 + kk);
            const v2f bb = *(const v2f*)(Wrow  + k0 + kk);
            acc0 = __builtin_amdgcn_wmma_f32_16x16x4_f32(
                       false, a0, false, bb, (short)0, acc0, false, false);
            acc1 = __builtin_amdgcn_wmma_f32_16x16x4_f32(
                       false, a1, false, bb, (short)0, acc1, false, false);
        }
    }

    // Epilogue: routing scale + bias. Block indices are uniform per 16-wide
    // tile (768 and 192 are multiples of 16, tiles never straddle).
    const int   k1   = nt / 48;                  // o / 768
    const int   k2   = (nt % 48) / 12;           // (o % 768) / 192
    const float bval = bias[ncol];
    const int   hi8  = (lane >> 4) * 8;

    #pragma unroll
    for (int r = 0; r < 8; ++r) {
        const int b0 = r + hi8;        // batch row of acc0[r]
        const int b1 = b0 + 16;        // batch row of acc1[r]
        float s0, s1;
        if (IS_H) {
            s0 = kx[b0 * NBLK + k1] * ky[b0 * NBLK + k2];
            s1 = kx[b1 * NBLK + k1] * ky[b1 * NBLK + k2];
        } else {
            s0 = ky[b0 * NBLK + k1];
            s1 = ky[b1 * NBLK + k1];
        }
        G[(size_t)b0 * G3H + ncol] = s0 * acc0[r] + bval;
        G[(size_t)b1 * G3H + ncol] = s1 * acc1[r] + bval;
    }
}

// ---------------------------------------------------------------------------
// Kernel 3: GRU gate combine -> hy
// ---------------------------------------------------------------------------
__global__ __launch_bounds__(256) void bsgru_combine(
    const float* __restrict__ ig,      // [32,3072]
    const float* __restrict__ hg,      // [32,3072]
    const float* __restrict__ hx,      // [32,1024]
    const float* __restrict__ kx,      // [32,4]
    const float* __restrict__ ky,      // [32,4]
    float* __restrict__ hy)            // [32,1024]
{
    const int idx = blockIdx.x * blockDim.x + threadIdx.x;
    if (idx >= BATCH * HID) return;
    const int b = idx >> 10;           // / HID
    const int h = idx & (HID - 1);

    const float* igr = ig + (size_t)b * G3H;
    const float* hgr = hg + (size_t)b * G3H;
    const float iz = igr[h],         hz = hgr[h];
    const float ir = igr[HID + h],   hr = hgr[HID + h];
    const float in = igr[2*HID + h], hn = hgr[2*HID + h];

    const float z  = 1.0f / (1.0f + __expf(-(iz + hz)));   // update gate
    const float rg = 1.0f / (1.0f + __expf(-(ir + hr)));   // reset gate
    const float ng = tanhf(in + hn * rg);                  // new gate

    const int   kb  = h >> 8;                              // h / BSIZE
    const float kxv = kx[b * NBLK + kb];
    const float kyv = ky[b * NBLK + kb];
    const float nk  = ng * kyv;
    hy[idx] = nk + z * (hx[idx] * kxv - nk);
}

// ---------------------------------------------------------------------------
// Host entry
// ---------------------------------------------------------------------------
extern "C" void kernel_launch(void* const* d_in, const int* in_sizes, int n_in,
                              void* d_out, int out_size, void* d_ws, size_t ws_size,
                              hipStream_t stream) {
    const float* input = (const float*)d_in[0];   // [32,512]
    const float* hx    = (const float*)d_in[1];   // [32,1024]
    const float* kx    = (const float*)d_in[2];   // [32,4]
    const float* wik   = (const float*)d_in[3];   // [512,4]
    const float* whk   = (const float*)d_in[4];   // [1024,4]
    const float* wih   = (const float*)d_in[5];   // [1,4,512,768]  -> flat [3072,512]
    const float* whh   = (const float*)d_in[6];   // [1,4,4,256,768]-> flat [3072,1024]
    const float* bik   = (const float*)d_in[7];   // [4]
    const float* bih   = (const float*)d_in[8];   // [3072]
    const float* bhh   = (const float*)d_in[9];   // [3072]

    float* out    = (float*)d_out;                // hy [32,1024] ++ ky [32,4]
    float* hy_out = out;
    float* ky_out = out + BATCH * HID;

    float* ws = (float*)d_ws;
    float* ky_ws = ws;                            // 128 floats
    float* ig_ws = ws + 128;                      // 32*3072
    float* hg_ws = ig_ws + BATCH * G3H;           // 32*3072

    bsgru_route<<<1, 128, 0, stream>>>(input, hx, wik, whk, bik, ky_ws, ky_out);

    // 192 wave-tiles each, 4 waves/block -> 48 blocks per GEMM
    bsgru_gemm<IN_SZ, false><<<48, 128, 0, stream>>>(
        input, wih, bih, nullptr, ky_ws, ig_ws);
    bsgru_gemm<HID, true><<<48, 128, 0, stream>>>(
        hx, whh, bhh, kx, ky_ws, hg_ws);

    bsgru_combine<<<(BATCH * HID + 255) / 256, 256, 0, stream>>>(
        ig_ws, hg_ws, hx, kx, ky_ws, hy_out);
}